// EMDLoss_51153060495391
// MI455X (gfx1250) — compile-verified
//
#include <hip/hip_runtime.h>
#include <math.h>

// ---------------------------------------------------------------------------
// Entropy-regularized OT (Sinkhorn) for B=16, N=2048, 3-D points, eps=0.02.
// Streaming (flash-attention-style) logsumexp over the 2048x2048 cost matrix.
// The Gram term x_i . y_j is computed per 16x16 tile with
// V_WMMA_F32_16X16X4_F32 (K=3 padded to 4); all affine terms of
// z_ij = (g_j - C_ij)/eps are folded into the WMMA C-operand so the MMA
// output is the LSE argument directly (in base-2 domain).
// Online LSE uses a single raw v_exp_f32 per element (arg = -|z-m| via
// free neg/abs src modifiers; underflow-flush is numerically harmless).
// ---------------------------------------------------------------------------

typedef __attribute__((ext_vector_type(2))) float v2f;
typedef __attribute__((ext_vector_type(8))) float v8f;

#define BATCH 16
#define NPTS  2048
#define BN    (BATCH * NPTS)       // 32768
#define TILES (NPTS / 16)          // 128 column tiles
#define WPB   8                    // waves per 256-thread block
#define EPS_R 0.02f
#define LOG2E 1.4426950408889634f
#define ZSCALE (50.0f * LOG2E)                 // (1/eps) * log2(e)
#define INV_ZSCALE (1.0f / (50.0f * 1.4426950408889634f))
#define LN2   0.69314718055994531f
#define LOGMU (-11.0f * 0.69314718055994531f)  // -ln(2048)
#define LOGMUNU_LOG2 (-22.0f)                  // (logmu+lognu)*log2e

static __device__ __forceinline__ float fast_exp2(float x) {
    return __builtin_amdgcn_exp2f(x);          // bare v_exp_f32
}

__global__ __launch_bounds__(256)
void init_kernel(float* __restrict__ f, float* __restrict__ g, float* __restrict__ out) {
    int idx = blockIdx.x * blockDim.x + threadIdx.x;
    if (idx < BN) { f[idx] = 0.0f; g[idx] = 0.0f; }
    if (idx == 0) out[0] = 0.0f;
}

__global__ __launch_bounds__(256)
void sq_kernel(const float* __restrict__ x, const float* __restrict__ y,
               float* __restrict__ xsq, float* __restrict__ ysq) {
    int idx = blockIdx.x * blockDim.x + threadIdx.x;
    if (idx < BN) {
        const float* p = x + (size_t)idx * 3;
        xsq[idx] = p[0]*p[0] + p[1]*p[1] + p[2]*p[2];
        const float* q = y + (size_t)idx * 3;
        ysq[idx] = q[0]*q[0] + q[1]*q[1] + q[2]*q[2];
    }
}

// pot_r_out[i] = eps * (logmu - logsumexp_j((pot_c[j] - C_ij)/eps))
__global__ __launch_bounds__(256)
void sinkhorn_update(const float* __restrict__ rows, const float* __restrict__ rsq,
                     const float* __restrict__ cols, const float* __restrict__ csq,
                     const float* __restrict__ pot_c, float* __restrict__ pot_r_out) {
    const int lane = threadIdx.x & 31;
    const int gw   = blockIdx.x * WPB + (threadIdx.x >> 5);  // 0..2047
    const int b    = gw / TILES;
    const int i0   = (gw % TILES) * 16;
    const int half = lane >> 4;
    const int l    = lane & 15;
    const size_t base = (size_t)b * NPTS;

    const float* rowb = rows + base * 3;     // uniform bases, 32-bit lane offsets
    const float* cb   = cols + base * 3;
    const float* cqb  = csq + base;
    const float* pcb  = pot_c + base;
    const float* rsb  = rsq + base;

    // A fragment (16x4 f32): lanes 0-15 hold (K0,K1), lanes 16-31 (K2,pad).
    // Pre-scale by 2/eps*log2e so the MMA produces base-2 LSE arguments.
    v2f a;
    {
        const int ri = (i0 + l) * 3;
        float r0 = rowb[ri], r1 = rowb[ri + 1], r2 = rowb[ri + 2];
        a.x = (half ? r2 : r0) * (2.0f * ZSCALE);
        a.y = half ? 0.0f : r1 * (2.0f * ZSCALE);
    }

    float rz[8], m[8], s[8];
#pragma unroll
    for (int r = 0; r < 8; ++r) {            // acc[r] = row r (low half) / r+8 (high half)
        rz[r] = rsb[i0 + half * 8 + r] * ZSCALE;
        m[r]  = -1e30f;
        s[r]  = 0.0f;
    }

    // software-pipelined column stream
    int j = l;
    float y0 = cb[3*j], y1 = cb[3*j + 1], y2 = cb[3*j + 2];
    float cq = cqb[j], pc = pcb[j];

    for (int jt = 0; jt < TILES; ++jt) {
        const int jn = (jt == TILES - 1) ? l : (j + 16);
        float n0 = cb[3*jn], n1 = cb[3*jn + 1], n2 = cb[3*jn + 2];
        float ncq = cqb[jn], npc = pcb[jn];

        v2f bb;                              // B fragment: 4x16 f32, same K split
        bb.x = half ? y2 : y0;
        bb.y = half ? 0.0f : y1;
        const float t0 = (pc - cq) * ZSCALE;
        v8f cin;
#pragma unroll
        for (int r = 0; r < 8; ++r) cin[r] = t0 - rz[r];

        // z[r] = (pot_c[j] - C_ij)/eps * log2e, directly from the MMA
        v8f z = __builtin_amdgcn_wmma_f32_16x16x4_f32(false, a, false, bb,
                                                      (short)0, cin, false, false);
#pragma unroll
        for (int r = 0; r < 8; ++r) {        // single-exp online LSE update
            float d  = z[r] - m[r];
            float e  = fast_exp2(-__builtin_fabsf(d));   // neg/abs are src modifiers
            float s1 = __builtin_fmaf(s[r], e, 1.0f);    // new max case
            float s2 = s[r] + e;                         // old max case
            s[r] = (d > 0.0f) ? s1 : s2;
            m[r] = fmaxf(m[r], z[r]);
        }
        y0 = n0; y1 = n1; y2 = n2; cq = ncq; pc = npc; j = jn;
    }

    // combine the 16 columns held by each 16-lane half (single exp per merge)
#pragma unroll
    for (int off = 1; off <= 8; off <<= 1) {
#pragma unroll
        for (int r = 0; r < 8; ++r) {
            float mo = __shfl_xor(m[r], off, 32);
            float so = __shfl_xor(s[r], off, 32);
            float d  = mo - m[r];
            float e  = fast_exp2(-__builtin_fabsf(d));
            float s1 = __builtin_fmaf(s[r], e, so);
            float s2 = __builtin_fmaf(so, e, s[r]);
            s[r] = (d > 0.0f) ? s1 : s2;
            m[r] = fmaxf(m[r], mo);
        }
    }
    if (l < 8) {
        float mf = 0.0f, sf = 0.0f;
#pragma unroll
        for (int r = 0; r < 8; ++r)          // static selects, no scratch spill
            if (l == r) { mf = m[r]; sf = s[r]; }
        float lse_nat = (mf + log2f(sf)) * LN2;
        pot_r_out[base + i0 + half * 8 + l] = EPS_R * (LOGMU - lse_nat);
    }
}

// out += (10/B) * sum_ij exp((f_i+g_j-C_ij)/eps + logmu+lognu) * C_ij
__global__ __launch_bounds__(256)
void cost_kernel(const float* __restrict__ x, const float* __restrict__ xsq,
                 const float* __restrict__ y, const float* __restrict__ ysq,
                 const float* __restrict__ f, const float* __restrict__ g,
                 float* __restrict__ out) {
    const int lane = threadIdx.x & 31;
    const int gw   = blockIdx.x * WPB + (threadIdx.x >> 5);
    const int b    = gw / TILES;
    const int i0   = (gw % TILES) * 16;
    const int half = lane >> 4;
    const int l    = lane & 15;
    const size_t base = (size_t)b * NPTS;

    const float* xb  = x + base * 3;
    const float* yb  = y + base * 3;
    const float* xqb = xsq + base;
    const float* yqb = ysq + base;
    const float* fb  = f + base;
    const float* gb  = g + base;

    v2f a;
    {
        const int ri = (i0 + l) * 3;
        float r0 = xb[ri], r1 = xb[ri + 1], r2 = xb[ri + 2];
        a.x = (half ? r2 : r0) * (2.0f * ZSCALE);
        a.y = half ? 0.0f : r1 * (2.0f * ZSCALE);
    }

    float u[8], fr[8];
#pragma unroll
    for (int r = 0; r < 8; ++r) {
        float fv = fb[i0 + half * 8 + r];
        float rs = xqb[i0 + half * 8 + r];
        u[r]  = (fv - rs) * ZSCALE;          // row part of the C-operand
        fr[r] = fv;
    }

    float acc_cost = 0.0f;
    int j = l;
    float y0 = yb[3*j], y1 = yb[3*j + 1], y2 = yb[3*j + 2];
    float cq = yqb[j], gc = gb[j];

    for (int jt = 0; jt < TILES; ++jt) {
        const int jn = (jt == TILES - 1) ? l : (j + 16);
        float n0 = yb[3*jn], n1 = yb[3*jn + 1], n2 = yb[3*jn + 2];
        float ncq = yqb[jn], ngc = gb[jn];

        v2f bb;
        bb.x = half ? y2 : y0;
        bb.y = half ? 0.0f : y1;
        const float v = (gc - cq) * ZSCALE + LOGMUNU_LOG2;
        v8f cin;
#pragma unroll
        for (int r = 0; r < 8; ++r) cin[r] = u[r] + v;

        // t[r] = (f_i+g_j-C_ij)/eps*log2e - 22, directly from the MMA
        v8f t = __builtin_amdgcn_wmma_f32_16x16x4_f32(false, a, false, bb,
                                                      (short)0, cin, false, false);
        const float h2b = gc - 22.0f * INV_ZSCALE;
        float hh[8];
#pragma unroll
        for (int r = 0; r < 8; ++r) hh[r] = fr[r] + h2b;   // f_i + g_j - 22/Z
#pragma unroll
        for (int r = 0; r < 8; ++r) {
            float C = __builtin_fmaf(-INV_ZSCALE, t[r], hh[r]);  // reconstruct C_ij
            float e = fast_exp2(t[r]);
            acc_cost = __builtin_fmaf(e, C, acc_cost);
        }
        y0 = n0; y1 = n1; y2 = n2; cq = ncq; gc = ngc; j = jn;
    }
#pragma unroll
    for (int off = 1; off <= 16; off <<= 1)
        acc_cost += __shfl_xor(acc_cost, off, 32);
    if (lane == 0)
        atomicAdd(out, acc_cost * (10.0f / (float)BATCH));
}

extern "C" void kernel_launch(void* const* d_in, const int* in_sizes, int n_in,
                              void* d_out, int out_size, void* d_ws, size_t ws_size,
                              hipStream_t stream) {
    (void)in_sizes; (void)n_in; (void)out_size; (void)ws_size;
    const float* x = (const float*)d_in[0];
    const float* y = (const float*)d_in[1];
    float* out = (float*)d_out;
    float* ws  = (float*)d_ws;
    float* xsq = ws;
    float* ysq = ws + BN;
    float* fpt = ws + 2 * BN;
    float* gpt = ws + 3 * BN;

    const int grid = (BATCH * TILES) / WPB;  // 256 blocks x 256 threads = 2048 waves

    init_kernel<<<BN / 256, 256, 0, stream>>>(fpt, gpt, out);
    sq_kernel<<<BN / 256, 256, 0, stream>>>(x, y, xsq, ysq);
    for (int it = 0; it < 20; ++it) {
        // f update reads g; g update reads the fresh f (matches reference order)
        sinkhorn_update<<<grid, 256, 0, stream>>>(x, xsq, y, ysq, gpt, fpt);
        sinkhorn_update<<<grid, 256, 0, stream>>>(y, ysq, x, xsq, fpt, gpt);
    }
    cost_kernel<<<grid, 256, 0, stream>>>(x, xsq, y, ysq, fpt, gpt, out);
}